// MultipleHeatmapLoss_30537217474919
// MI455X (gfx1250) — compile-verified
//
#include <hip/hip_runtime.h>

// MultipleHeatmapLoss: loss = sum((out-tgt)^2) / (H*W) / B * WEIGHT
// Shapes: [64, 17, 256, 256] f32.  N = 71,303,168 elements per input.
// Memory-bound: ~570 MB streamed @ 23.3 TB/s -> ~24.5 us floor.

typedef __attribute__((ext_vector_type(2))) float v2f;
typedef __attribute__((ext_vector_type(4))) float v4f;
typedef __attribute__((ext_vector_type(8))) float v8f;

#define RED_BLOCKS  2048
#define RED_THREADS 256
#define WAVES_PER_BLOCK (RED_THREADS / 32)

static constexpr long long N_ELEM = 64LL * 17LL * 256LL * 256LL;   // 71,303,168
static constexpr long long N_VEC4 = N_ELEM / 4;                     // 17,825,792
// WEIGHT / (H*W) / batches = 1 / (256*256*64) = 2^-22 (exact in f32)
static constexpr float FINAL_SCALE = 1.0f / 4194304.0f;

__global__ __launch_bounds__(RED_THREADS)
void hm_loss_partials(const float* __restrict__ out_p,
                      const float* __restrict__ tgt_p,
                      float* __restrict__ partials)
{
    const v4f* __restrict__ o4 = (const v4f*)out_p;
    const v4f* __restrict__ t4 = (const v4f*)tgt_p;

    const long long tid    = (long long)blockIdx.x * RED_THREADS + threadIdx.x;
    const long long stride = (long long)gridDim.x * RED_THREADS;

    // Two independent accumulators per lane (ILP + feeds the 16x4 A-matrix).
    float acc0 = 0.0f, acc1 = 0.0f;

    for (long long i = tid; i < N_VEC4; i += stride) {
        // Streaming (non-temporal) 128-bit loads: don't pollute L2 with a
        // 570 MB one-pass stream.
        v4f a = __builtin_nontemporal_load(o4 + i);
        v4f b = __builtin_nontemporal_load(t4 + i);
        float dx = a.x - b.x;
        float dy = a.y - b.y;
        float dz = a.z - b.z;
        float dw = a.w - b.w;
        acc0 = fmaf(dx, dx, acc0);
        acc1 = fmaf(dy, dy, acc1);
        acc0 = fmaf(dz, dz, acc0);
        acc1 = fmaf(dw, dw, acc1);
    }

    // ---- Wave-level reduction via V_WMMA_F32_16X16X4_F32 -------------------
    // A (16x4 f32, 2 VGPRs/lane) = the wave's 64 accumulators.
    // B (4x16 f32) = all ones  =>  D[m][n] = sum_k A[m][k]  (every column equal).
    // Per-lane sum of the 8 D VGPRs = sum of rows M=0..7 (lanes 0-15) or
    // M=8..15 (lanes 16-31); one shfl_xor(16) yields the full wave sum.
    v2f amat; amat.x = acc0; amat.y = acc1;
    v2f bmat; bmat.x = 1.0f; bmat.y = 1.0f;
    v8f cmat = {};
    cmat = __builtin_amdgcn_wmma_f32_16x16x4_f32(
        /*neg_a=*/false, amat, /*neg_b=*/false, bmat,
        /*c_mod=*/(short)0, cmat, /*reuse_a=*/false, /*reuse_b=*/false);

    float s = cmat[0] + cmat[1] + cmat[2] + cmat[3]
            + cmat[4] + cmat[5] + cmat[6] + cmat[7];
    s += __shfl_xor(s, 16, 32);   // combine M=0..7 with M=8..15 halves

    // ---- Cross-wave reduction in LDS --------------------------------------
    __shared__ float wave_sums[WAVES_PER_BLOCK];
    const int lane = threadIdx.x & 31;
    const int wave = threadIdx.x >> 5;
    if (lane == 0) wave_sums[wave] = s;
    __syncthreads();

    if (wave == 0) {
        float v = (lane < WAVES_PER_BLOCK) ? wave_sums[lane] : 0.0f;
        v += __shfl_xor(v, 4, 32);
        v += __shfl_xor(v, 2, 32);
        v += __shfl_xor(v, 1, 32);
        if (lane == 0) partials[blockIdx.x] = v;
    }
}

__global__ __launch_bounds__(RED_THREADS)
void hm_loss_final(const float* __restrict__ partials, float* __restrict__ out)
{
    __shared__ float wave_sums[WAVES_PER_BLOCK];

    float v = 0.0f;
    for (int i = threadIdx.x; i < RED_BLOCKS; i += RED_THREADS)
        v += partials[i];

    v += __shfl_xor(v, 16, 32);
    v += __shfl_xor(v,  8, 32);
    v += __shfl_xor(v,  4, 32);
    v += __shfl_xor(v,  2, 32);
    v += __shfl_xor(v,  1, 32);

    const int lane = threadIdx.x & 31;
    const int wave = threadIdx.x >> 5;
    if (lane == 0) wave_sums[wave] = v;
    __syncthreads();

    if (threadIdx.x == 0) {
        float t = 0.0f;
        for (int i = 0; i < WAVES_PER_BLOCK; ++i) t += wave_sums[i];
        out[0] = t * FINAL_SCALE;
    }
}

extern "C" void kernel_launch(void* const* d_in, const int* in_sizes, int n_in,
                              void* d_out, int out_size, void* d_ws, size_t ws_size,
                              hipStream_t stream)
{
    (void)in_sizes; (void)n_in; (void)out_size; (void)ws_size;
    const float* out_p = (const float*)d_in[0];   // "output"
    const float* tgt_p = (const float*)d_in[1];   // "target"
    float* partials = (float*)d_ws;               // RED_BLOCKS floats (8 KB)
    float* loss     = (float*)d_out;              // 1 float

    hm_loss_partials<<<RED_BLOCKS, RED_THREADS, 0, stream>>>(out_p, tgt_p, partials);
    hm_loss_final<<<1, RED_THREADS, 0, stream>>>(partials, loss);
}